// GraphEncoder_42631845380409
// MI455X (gfx1250) — compile-verified
//
#include <hip/hip_runtime.h>
#include <hip/hip_bf16.h>

#define N_NODES_C 100000
#define N_EDGES_C 1600000
#define IN_DIM_C  128
#define HID_C     32
#define BN_EPS_C  1e-5f

typedef float v2f __attribute__((ext_vector_type(2)));
typedef float v8f __attribute__((ext_vector_type(8)));

// ---------------------------------------------------------------------------
// Utility fills
// ---------------------------------------------------------------------------
__global__ void fill_f32(float* __restrict__ p, int n, float v) {
    int i = blockIdx.x * blockDim.x + threadIdx.x;
    if (i < n) p[i] = v;
}

__global__ void fill_f32_vec4(float4* __restrict__ p, int n4, float v) {
    int i = blockIdx.x * blockDim.x + threadIdx.x;
    if (i < n4) p[i] = make_float4(v, v, v, v);
}

// out[node*HID + f] = b[f], vectorized: 8 float4 per node row (HID==32)
__global__ void init_out_bias(float4* __restrict__ out, const float4* __restrict__ b4,
                              int n_nodes) {
    int i = blockIdx.x * blockDim.x + threadIdx.x;
    int total = n_nodes * (HID_C / 4);
    if (i < total) out[i] = b4[i & (HID_C / 4 - 1)];
}

// ---------------------------------------------------------------------------
// Degree (self-loop baked in via deg init = 1.0) and d^{-1/2}
// ---------------------------------------------------------------------------
__global__ void degree_kernel(const int* __restrict__ dst, float* __restrict__ deg, int n_edges) {
    int i = blockIdx.x * blockDim.x + threadIdx.x;
    if (i < n_edges) atomicAdd(deg + dst[i], 1.0f);
}

__global__ void rsqrt_kernel(float* __restrict__ deg, int n) {
    int i = blockIdx.x * blockDim.x + threadIdx.x;
    if (i < n) deg[i] = rsqrtf(deg[i]);   // deg >= 1 guaranteed (self loop)
}

// ---------------------------------------------------------------------------
// FP32 WMMA GEMM: C[M x 32] = A[M x K] * B[K x 32],  K compile-time.
// One wave computes a full 16x32 output strip: two 16x16 tiles sharing one
// A fetch (two accumulators). 8 waves per block, one row-tile per wave.
//
// V_WMMA_F32_16X16X4_F32 layouts (ISA 7.12.2, 32-bit A 16x4):
//   A: lanes 0-15 hold K={k,k+1} (VGPR0,VGPR1), lanes 16-31 hold K={k+2,k+3};
//      M = lane%16.  B mirrors with N = lane%16.
//   C/D: VGPR j -> row (j + 8*(lane>=16)), col lane%16.
// ---------------------------------------------------------------------------
template <int K>
__global__ void gemm_wmma_f32(const float* __restrict__ A, const float* __restrict__ B,
                              float* __restrict__ C, int n_tiles) {
    const int lane = threadIdx.x & 31;
    const int wid  = threadIdx.x >> 5;                 // 0..7
    const int tile = blockIdx.x * 8 + wid;
    if (tile >= n_tiles) return;                       // wave-uniform guard

    const int m    = lane & 15;
    const int half = lane >> 4;                        // 0 or 1
    const int row0 = tile * 16;

    const float* __restrict__ arow = A + (size_t)(row0 + m) * K;
    v8f acc0 = {};                                     // cols 0..15
    v8f acc1 = {};                                     // cols 16..31

#pragma unroll 8
    for (int k = 0; k < K; k += 4) {
        const int ka = k + half * 2;
        v2f a, b0, b1;
        a.x  = arow[ka];
        a.y  = arow[ka + 1];
        b0.x = B[(size_t)ka * HID_C + m];
        b0.y = B[(size_t)(ka + 1) * HID_C + m];
        b1.x = B[(size_t)ka * HID_C + 16 + m];
        b1.y = B[(size_t)(ka + 1) * HID_C + 16 + m];
        acc0 = __builtin_amdgcn_wmma_f32_16x16x4_f32(
            false, a, false, b0, (short)0, acc0, false, false);
        acc1 = __builtin_amdgcn_wmma_f32_16x16x4_f32(
            false, a, false, b1, (short)0, acc1, false, false);
    }

    float* __restrict__ crow = C + (size_t)(row0 + half * 8) * HID_C + m;
#pragma unroll
    for (int j = 0; j < 8; ++j) {
        crow[(size_t)j * HID_C]      = acc0[j];
        crow[(size_t)j * HID_C + 16] = acc1[j];
    }
}

// ---------------------------------------------------------------------------
// Edge + self-loop scatter:  out[dst] += h[src] * dinv[src]*dinv[dst]
// One wave per item, lane = feature (HID == 32 == wave32).
// ---------------------------------------------------------------------------
__global__ void scatter_conv(const float* __restrict__ h, const int* __restrict__ src,
                             const int* __restrict__ dst, const float* __restrict__ dinv,
                             float* __restrict__ out, int n_edges, int n_nodes) {
    const int item = blockIdx.x * (blockDim.x >> 5) + (threadIdx.x >> 5);
    const int f    = threadIdx.x & 31;
    const int total = n_edges + n_nodes;
    if (item >= total) return;

    int s, d;
    float nrm;
    if (item < n_edges) {
        s = src[item];
        d = dst[item];
        nrm = dinv[s] * dinv[d];
    } else {                       // self loop
        s = d = item - n_edges;
        float di = dinv[s];
        nrm = di * di;
    }
    const float v = h[(size_t)s * HID_C + f] * nrm;
    atomicAdd(out + (size_t)d * HID_C + f, v);
}

// ---------------------------------------------------------------------------
// BatchNorm statistics: per-feature sum and sum-of-squares.
// 256 threads = 8 node-groups x 32 features; LDS reduce then global atomics.
// ---------------------------------------------------------------------------
__global__ void bn_stats(const float* __restrict__ h, float* __restrict__ sums, int n_nodes) {
    __shared__ float ssum[256];
    __shared__ float ssq[256];
    const int f   = threadIdx.x & 31;
    const int grp = threadIdx.x >> 5;                 // 0..7
    const int stride = gridDim.x * 8;

    float s = 0.0f, q = 0.0f;
    for (int node = blockIdx.x * 8 + grp; node < n_nodes; node += stride) {
        const float v = h[(size_t)node * HID_C + f];
        s += v;
        q += v * v;
    }
    ssum[threadIdx.x] = s;
    ssq[threadIdx.x]  = q;
    __syncthreads();
    if (threadIdx.x < 32) {
#pragma unroll
        for (int g = 1; g < 8; ++g) {
            s += ssum[g * 32 + f];
            q += ssq[g * 32 + f];
        }
        atomicAdd(&sums[f], s);
        atomicAdd(&sums[32 + f], q);
    }
}

// sums[64+f] = scale, sums[96+f] = shift  (single 32-thread block)
__global__ void bn_finalize(float* __restrict__ sums, const float* __restrict__ gamma,
                            const float* __restrict__ beta, int n_nodes) {
    const int f = threadIdx.x;
    const float invn = 1.0f / (float)n_nodes;
    const float mean = sums[f] * invn;
    const float var  = sums[32 + f] * invn - mean * mean;   // biased, matches reference
    const float sc   = gamma[f] * rsqrtf(var + BN_EPS_C);
    sums[64 + f] = sc;
    sums[96 + f] = beta[f] - mean * sc;
}

// hbn = relu(agg * scale + shift), float4-vectorized (HID % 4 == 0)
__global__ void bn_apply_relu(const float4* __restrict__ agg, const float* __restrict__ sums,
                              float4* __restrict__ out, int n_nodes) {
    int i = blockIdx.x * blockDim.x + threadIdx.x;
    const int total = n_nodes * (HID_C / 4);
    if (i < total) {
        const int f4 = (i & (HID_C / 4 - 1)) * 4;
        const float4 v = agg[i];
        float4 r;
        r.x = fmaxf(v.x * sums[64 + f4 + 0] + sums[96 + f4 + 0], 0.0f);
        r.y = fmaxf(v.y * sums[64 + f4 + 1] + sums[96 + f4 + 1], 0.0f);
        r.z = fmaxf(v.z * sums[64 + f4 + 2] + sums[96 + f4 + 2], 0.0f);
        r.w = fmaxf(v.w * sums[64 + f4 + 3] + sums[96 + f4 + 3], 0.0f);
        out[i] = r;
    }
}

// ---------------------------------------------------------------------------
// Host-side orchestration
// ---------------------------------------------------------------------------
extern "C" void kernel_launch(void* const* d_in, const int* in_sizes, int n_in,
                              void* d_out, int out_size, void* d_ws, size_t ws_size,
                              hipStream_t stream) {
    (void)in_sizes; (void)n_in; (void)out_size; (void)ws_size;

    const float* x     = (const float*)d_in[0];            // [N, 128]
    const int*   edges = (const int*)d_in[1];              // [2, E]
    const float* W1    = (const float*)d_in[2];            // [128, 32]
    // d_in[3] = b1: mathematically cancelled by BatchNorm mean-subtraction.
    const float* gamma = (const float*)d_in[4];            // [32]
    const float* beta  = (const float*)d_in[5];            // [32]
    const float* W2    = (const float*)d_in[6];            // [32, 32]
    const float* b2    = (const float*)d_in[7];            // [32]

    const int N = N_NODES_C;
    const int E = N_EDGES_C;
    const int* src = edges;
    const int* dst = edges + E;

    float* ws   = (float*)d_ws;
    float* deg  = ws;                                      // N floats (becomes dinv)
    float* bufA = deg + N;                                 // N*32  (h1, later hbn)
    float* bufB = bufA + (size_t)N * HID_C;                // N*32  (agg1, later g2)
    float* sums = bufB + (size_t)N * HID_C;                // 128 floats

    float* out = (float*)d_out;

    const int TPB = 256;
    const int NH  = N * HID_C;
    const int NH4 = NH / 4;
    const int n_tiles = N / 16;                            // 6250

    // --- init ---
    fill_f32<<<(N + TPB - 1) / TPB, TPB, 0, stream>>>(deg, N, 1.0f);             // self loop
    fill_f32_vec4<<<(NH4 + TPB - 1) / TPB, TPB, 0, stream>>>((float4*)bufB, NH4, 0.0f);
    fill_f32<<<1, 128, 0, stream>>>(sums, 128, 0.0f);                            // bn sums
    init_out_bias<<<(NH4 + TPB - 1) / TPB, TPB, 0, stream>>>(
        (float4*)out, (const float4*)b2, N);                                     // out = b2

    // --- normalization coefficients ---
    degree_kernel<<<(E + TPB - 1) / TPB, TPB, 0, stream>>>(dst, deg, E);
    rsqrt_kernel<<<(N + TPB - 1) / TPB, TPB, 0, stream>>>(deg, N);               // deg -> dinv

    // --- layer 1: h1 = x @ W1 (WMMA f32), then scatter ---
    gemm_wmma_f32<IN_DIM_C><<<(n_tiles + 7) / 8, 256, 0, stream>>>(x, W1, bufA, n_tiles);
    {
        const int items = E + N;                           // edges + self loops
        const int ipb = TPB / 32;                          // 8 items per block
        scatter_conv<<<(items + ipb - 1) / ipb, TPB, 0, stream>>>(
            bufA, src, dst, deg, bufB, E, N);
    }

    // --- BatchNorm (training stats) + ReLU ---
    bn_stats<<<512, 256, 0, stream>>>(bufB, sums, N);
    bn_finalize<<<1, 32, 0, stream>>>(sums, gamma, beta, N);
    bn_apply_relu<<<(NH4 + TPB - 1) / TPB, TPB, 0, stream>>>(
        (const float4*)bufB, sums, (float4*)bufA, N);

    // --- layer 2: g2 = hbn @ W2 (WMMA f32), then scatter into d_out ---
    gemm_wmma_f32<HID_C><<<(n_tiles + 7) / 8, 256, 0, stream>>>(bufA, W2, bufB, n_tiles);
    {
        const int items = E + N;
        const int ipb = TPB / 32;
        scatter_conv<<<(items + ipb - 1) / ipb, TPB, 0, stream>>>(
            bufB, src, dst, deg, out, E, N);
    }
}